// GraphSAGE_66511863546568
// MI455X (gfx1250) — compile-verified
//
#include <hip/hip_runtime.h>

// ---------------------------------------------------------------------------
// GraphSAGE (2-layer, mean aggr) on MI455X / gfx1250.
// Trick: lin_l(mean(x_j)) == mean(lin_l(x_j)), so project BEFORE the edge
// aggregation. The only heavy GEMM ([100000x500]x[500x16]) is done with
// V_WMMA_F32_16X16X4_F32 (fp32 WMMA), B staged in LDS. Edge aggregation uses
// hardware float atomics on L2-resident 8/10-wide accumulators.
// ---------------------------------------------------------------------------

typedef __attribute__((ext_vector_type(2))) float v2f;
typedef __attribute__((ext_vector_type(8))) float v8f;

#define N_NODES 100000
#define F_IN    500
#define HID     8
#define NCLS    10
#define NCOL    16   // layer-1 projection: [y_l (8) | y_r (8)]
#define ZCOL    20   // layer-2 projection: [z_l (10) | z_r (10)]
#define N_TILES (N_NODES / 16)   // 6250, exact

// ---- Kernel 1: y = x @ [W1_l ; W1_r]^T via f32 WMMA ------------------------
// One wave computes a 16(nodes) x 16(cols) tile. K loop: 125 steps of K=4.
__global__ __launch_bounds__(256) void proj1_wmma(
    const float* __restrict__ x,
    const float* __restrict__ W1l,   // [8][500]
    const float* __restrict__ W1r,   // [8][500]
    float* __restrict__ y)           // [N][16]
{
    __shared__ float Wlds[NCOL * F_IN];   // 32 KB: rows 0..7 = W1l, 8..15 = W1r
    for (int i = threadIdx.x; i < NCOL * F_IN; i += 256)
        Wlds[i] = (i < HID * F_IN) ? W1l[i] : W1r[i - HID * F_IN];
    __syncthreads();

    const int wave = threadIdx.x >> 5;
    const int lane = threadIdx.x & 31;
    const int half = lane >> 4;       // selects K pair (A/B) and M half (C/D)
    const int n    = lane & 15;       // A: row M; B/D: column N
    const int tile = blockIdx.x * 8 + wave;
    if (tile >= N_TILES) return;      // wave-uniform guard (EXEC all-1 for WMMA)
    const int rbase = tile * 16;

    // A layout (16x4 f32): lane m=lane&15; VGPR0/1 = K = 2*half + {0,1}
    const float* arow = x    + (size_t)(rbase + n) * F_IN + 2 * half;
    // B layout (4x16 f32): lane n=lane&15; VGPR0/1 = K = 2*half + {0,1}
    const float* brow = Wlds + (size_t)n * F_IN + 2 * half;

    v8f acc = {0.f, 0.f, 0.f, 0.f, 0.f, 0.f, 0.f, 0.f};
    for (int kb = 0; kb < F_IN; kb += 4) {
        v2f a = *(const v2f*)(arow + kb);   // global_load_b64 (8B aligned)
        v2f b = *(const v2f*)(brow + kb);   // ds_load_b64     (8B aligned)
        acc = __builtin_amdgcn_wmma_f32_16x16x4_f32(
            /*neg_a=*/false, a, /*neg_b=*/false, b,
            /*c_mod=*/(short)0, acc, /*reuse_a=*/false, /*reuse_b=*/false);
    }

    // C/D layout: VGPR r -> row M = r + 8*half, column = n
    float* yo = y + (size_t)rbase * NCOL + n;
#pragma unroll
    for (int r = 0; r < 8; ++r)
        yo[(size_t)(r + 8 * half) * NCOL] = acc[r];
}

// ---- Kernel 2: layer-1 edge aggregation + degree count ---------------------
__global__ __launch_bounds__(256) void edge_agg1(
    const long long* __restrict__ ei, long long E,
    const float* __restrict__ y,      // [N][16], cols 0..7 = y_l
    float* __restrict__ agg,          // [N][8]
    float* __restrict__ cnt)          // [N]
{
    long long e = (long long)blockIdx.x * blockDim.x + threadIdx.x;
    if (e >= E) return;
    const int s = (int)ei[e];
    const int d = (int)ei[E + e];
    const float* ys = y + (size_t)s * NCOL;
    float4 v0 = *(const float4*)(ys);
    float4 v1 = *(const float4*)(ys + 4);
    float* ad = agg + (size_t)d * HID;
    unsafeAtomicAdd(ad + 0, v0.x); unsafeAtomicAdd(ad + 1, v0.y);
    unsafeAtomicAdd(ad + 2, v0.z); unsafeAtomicAdd(ad + 3, v0.w);
    unsafeAtomicAdd(ad + 4, v1.x); unsafeAtomicAdd(ad + 5, v1.y);
    unsafeAtomicAdd(ad + 6, v1.z); unsafeAtomicAdd(ad + 7, v1.w);
    unsafeAtomicAdd(cnt + d, 1.0f);
}

// ---- Kernel 3: mean + bias + ELU fused with layer-2 projection -------------
__global__ __launch_bounds__(256) void combine1(
    const float* __restrict__ agg,    // [N][8]
    const float* __restrict__ cnt,    // [N]
    const float* __restrict__ y,      // [N][16], cols 8..15 = y_r
    const float* __restrict__ b1,     // [8]
    const float* __restrict__ W2l,    // [10][8]
    const float* __restrict__ W2r,    // [10][8]
    float* __restrict__ z)            // [N][20] = [z_l(10) | z_r(10)]
{
    int i = blockIdx.x * blockDim.x + threadIdx.x;
    if (i >= N_NODES) return;
    const float inv = 1.0f / fmaxf(cnt[i], 1.0f);
    const float* ag = agg + (size_t)i * HID;
    const float* yr = y + (size_t)i * NCOL + HID;
    float h[HID];
#pragma unroll
    for (int k = 0; k < HID; ++k) {
        float v = ag[k] * inv + b1[k] + yr[k];
        h[k] = (v > 0.0f) ? v : (__expf(v) - 1.0f);   // ELU(alpha=1)
    }
    float* zi = z + (size_t)i * ZCOL;
#pragma unroll
    for (int j = 0; j < NCLS; ++j) {
        float sl = 0.0f, sr = 0.0f;
#pragma unroll
        for (int k = 0; k < HID; ++k) {
            sl += W2l[j * HID + k] * h[k];
            sr += W2r[j * HID + k] * h[k];
        }
        zi[j]        = sl;
        zi[NCLS + j] = sr;
    }
}

// ---- Kernel 4: layer-2 edge aggregation ------------------------------------
__global__ __launch_bounds__(256) void edge_agg2(
    const long long* __restrict__ ei, long long E,
    const float* __restrict__ z,      // [N][20], cols 0..9 = z_l
    float* __restrict__ agg2)         // [N][10]
{
    long long e = (long long)blockIdx.x * blockDim.x + threadIdx.x;
    if (e >= E) return;
    const int s = (int)ei[e];
    const int d = (int)ei[E + e];
    const float* zs = z + (size_t)s * ZCOL;
    float* ad = agg2 + (size_t)d * NCLS;
#pragma unroll
    for (int k = 0; k < NCLS; ++k)
        unsafeAtomicAdd(ad + k, zs[k]);
}

// ---- Kernel 5: mean + bias + log_softmax -----------------------------------
__global__ __launch_bounds__(256) void final_lsm(
    const float* __restrict__ agg2,   // [N][10]
    const float* __restrict__ cnt,    // [N]
    const float* __restrict__ z,      // [N][20], cols 10..19 = z_r
    const float* __restrict__ b2,     // [10]
    float* __restrict__ out)          // [N][10]
{
    int i = blockIdx.x * blockDim.x + threadIdx.x;
    if (i >= N_NODES) return;
    const float inv = 1.0f / fmaxf(cnt[i], 1.0f);
    const float* ag = agg2 + (size_t)i * NCLS;
    const float* zr = z + (size_t)i * ZCOL + NCLS;
    float v[NCLS];
    float m = -3.402823466e38f;
#pragma unroll
    for (int j = 0; j < NCLS; ++j) {
        v[j] = ag[j] * inv + b2[j] + zr[j];
        m = fmaxf(m, v[j]);
    }
    float s = 0.0f;
#pragma unroll
    for (int j = 0; j < NCLS; ++j) s += __expf(v[j] - m);
    const float lse = m + __logf(s);
    float* oi = out + (size_t)i * NCLS;
#pragma unroll
    for (int j = 0; j < NCLS; ++j) oi[j] = v[j] - lse;
}

// ---------------------------------------------------------------------------
extern "C" void kernel_launch(void* const* d_in, const int* in_sizes, int n_in,
                              void* d_out, int out_size, void* d_ws, size_t ws_size,
                              hipStream_t stream) {
    const float*     x   = (const float*)d_in[0];
    const long long* ei  = (const long long*)d_in[1];  // int64 [2][E]
    const float*     W1l = (const float*)d_in[2];
    const float*     b1l = (const float*)d_in[3];
    const float*     W1r = (const float*)d_in[4];
    const float*     W2l = (const float*)d_in[5];
    const float*     b2l = (const float*)d_in[6];
    const float*     W2r = (const float*)d_in[7];
    float*           out = (float*)d_out;

    const long long E = (long long)in_sizes[1] / 2;

    // Workspace layout (floats): y[16N] | z[20N] | cnt[N] | agg1[8N] | agg2[10N]
    float* ws   = (float*)d_ws;
    float* y    = ws;
    float* z    = y   + (size_t)16 * N_NODES;
    float* cnt  = z   + (size_t)20 * N_NODES;
    float* agg1 = cnt + (size_t)N_NODES;
    float* agg2 = agg1 + (size_t)8 * N_NODES;

    // Zero the accumulator region (cnt|agg1|agg2 contiguous) every call.
    hipMemsetAsync(cnt, 0, (size_t)(1 + 8 + 10) * N_NODES * sizeof(float), stream);

    const int gemm_blocks = (N_TILES + 7) / 8;              // 8 waves/block
    proj1_wmma<<<gemm_blocks, 256, 0, stream>>>(x, W1l, W1r, y);

    const int eb = (int)((E + 255) / 256);
    edge_agg1<<<eb, 256, 0, stream>>>(ei, E, y, agg1, cnt);

    const int nb = (N_NODES + 255) / 256;
    combine1<<<nb, 256, 0, stream>>>(agg1, cnt, y, b1l, W2l, W2r, z);

    edge_agg2<<<eb, 256, 0, stream>>>(ei, E, z, agg2);

    final_lsm<<<nb, 256, 0, stream>>>(agg2, cnt, z, b2l, out);
}